// SampleLayer_59966333387266
// MI455X (gfx1250) — compile-verified
//
#include <hip/hip_runtime.h>

// Problem constants (fixed by the harness's setup_inputs()).
constexpr int Bn      = 16;
constexpr int Hs      = 640;
constexpr int Ws      = 640;
constexpr int Cc      = 3;
constexpr int Nn      = 512 * 512;        // pixels per batch
constexpr long PIX    = (long)Bn * Nn;    // 4,194,304 total pixels
constexpr int  PPT    = 4;                // pixels per thread
constexpr int  BLOCK  = 256;

typedef float v4f __attribute__((ext_vector_type(4)));

__device__ __forceinline__ void sample_one(const float* __restrict__ srcB,
                                           float fr, float fc,
                                           float& r0, float& r1, float& r2) {
    // trunc-toward-zero of (x + 0.5), matching jnp.trunc(idx+0.5).astype(int32)
    int ir = (int)(fr + 0.5f);
    int ic = (int)(fc + 0.5f);
    bool valid = (ir >= 0) & (ic >= 0) & (ir < Hs) & (ic < Ws);
    int irc = min(max(ir, 0), Hs - 1);
    int icc = min(max(ic, 0), Ws - 1);
    const float* sp = srcB + ((size_t)irc * Ws + icc) * Cc;
    // 3 adjacent dword loads -> compiler merges to global_load_b96/b64+b32
    float a = sp[0];
    float b = sp[1];
    float c = sp[2];
    r0 = valid ? a : 0.0f;   // v_cndmask, branch-free, EXEC stays full
    r1 = valid ? b : 0.0f;
    r2 = valid ? c : 0.0f;
}

__global__ __launch_bounds__(BLOCK) void nn_gather_resample(
    const float* __restrict__ idx,     // [B, 2, N]
    const float* __restrict__ src,     // [B, Hs, Ws, C]
    float* __restrict__ out)           // [B, N, C]
{
    long t   = (long)blockIdx.x * blockDim.x + threadIdx.x;
    long pp0 = t * PPT;                // first pixel of this thread's group of 4
    if (pp0 >= PIX) return;

    int b = (int)(pp0 / Nn);
    int n = (int)(pp0 - (long)b * Nn); // multiple of 4 (4 | N), stays in-batch

    const float* rowp = idx + (size_t)b * 2 * Nn + n;  // row-coord plane
    const float* colp = rowp + Nn;                     // col-coord plane

    // Streamed once -> non-temporal b128 loads (TH=NT), 16B aligned.
    v4f fr = __builtin_nontemporal_load((const v4f*)rowp);
    v4f fc = __builtin_nontemporal_load((const v4f*)colp);

    const float* srcB = src + (size_t)b * Hs * Ws * Cc;

    float o[12];
    sample_one(srcB, fr.x, fc.x, o[0], o[1],  o[2]);
    sample_one(srcB, fr.y, fc.y, o[3], o[4],  o[5]);
    sample_one(srcB, fr.z, fc.z, o[6], o[7],  o[8]);
    sample_one(srcB, fr.w, fc.w, o[9], o[10], o[11]);

    // 48 bytes out per thread; base = 48*t -> 16B aligned. Write-once -> NT stores.
    v4f* op = (v4f*)(out + pp0 * Cc);
    v4f s0 = { o[0], o[1], o[2],  o[3]  };
    v4f s1 = { o[4], o[5], o[6],  o[7]  };
    v4f s2 = { o[8], o[9], o[10], o[11] };
    __builtin_nontemporal_store(s0, op + 0);
    __builtin_nontemporal_store(s1, op + 1);
    __builtin_nontemporal_store(s2, op + 2);
}

extern "C" void kernel_launch(void* const* d_in, const int* in_sizes, int n_in,
                              void* d_out, int out_size, void* d_ws, size_t ws_size,
                              hipStream_t stream) {
    const float* idx = (const float*)d_in[0];   // [16, 2, 262144] f32
    const float* src = (const float*)d_in[1];   // [16, 640, 640, 3] f32
    float* out = (float*)d_out;                 // [16, 512, 512, 3] f32
    (void)in_sizes; (void)n_in; (void)d_ws; (void)ws_size; (void)out_size;

    long threads = PIX / PPT;                   // 1,048,576
    int  grid    = (int)((threads + BLOCK - 1) / BLOCK);  // 4096
    nn_gather_resample<<<grid, BLOCK, 0, stream>>>(idx, src, out);
}